// Despawn_65541200937648
// MI455X (gfx1250) — compile-verified
//
#include <hip/hip_runtime.h>
#include <hip/hip_bf16.h>

typedef __attribute__((ext_vector_type(2))) float    v2f;
typedef __attribute__((ext_vector_type(8))) float    v8f;
typedef __attribute__((ext_vector_type(4))) int      i32x4;
typedef __attribute__((ext_vector_type(8))) int      i32x8;
typedef __attribute__((ext_vector_type(4))) unsigned u32x4;
typedef __attribute__((address_space(1))) i32x4* gptr4;
typedef __attribute__((address_space(3))) i32x4* lptr4;

#define LEVELS 12
#define KTAPS  16
#define M_IN   16000
#define LPAD   16384
#define PL     192
#define HALO   8

__device__ __forceinline__ float sigm(float z) { return 1.0f / (1.0f + __expf(-z)); }
__device__ __forceinline__ int   iclamp(int v, int lo, int hi) {
  return v < lo ? lo : (v > hi ? hi : v);
}

// Build TDM D# groups for a contiguous 1 x N f32 tile (global <-> LDS).
__device__ __forceinline__ u32x4 tdm_g0(const void* gaddr, unsigned lds_off) {
  unsigned long long ga = (unsigned long long)(uintptr_t)gaddr;
  u32x4 g0;
  g0.x = 1u;                                   // count=1 valid, user mode
  g0.y = lds_off;                              // lds_addr (bytes)
  g0.z = (unsigned)ga;                         // global_addr[31:0]
  g0.w = (unsigned)(ga >> 32) | (2u << 30);    // global_addr[56:32] | type=2
  return g0;
}
__device__ __forceinline__ i32x8 tdm_g1(void) {
  i32x8 g1;
  g1[0] = 0x00020000;            // wg_mask=0 | data_size=2 (4B)
  g1[1] = (int)(16000u << 16);   // tensor_dim0[15:0] << 16 (=16000)
  g1[2] = 0x00010000;            // tensor_dim0[31:16]=0 | tensor_dim1=1
  g1[3] = (int)(16000u << 16);   // tile_dim0 = 16000
  g1[4] = 1;                     // tile_dim1 = 1, tile_dim2 = 0
  g1[5] = 16000;                 // tensor_dim0_stride[31:0]
  g1[6] = 0;                     // stride0[47:32] | stride1[15:0]
  g1[7] = 0;
  return g1;
}

__global__ void Despawn_zero_kernel(float* p) { *p = 0.0f; }

__global__ __launch_bounds__(256) void Despawn_dwt_kernel(
    const float* __restrict__ x, const float* __restrict__ scaling,
    const float* __restrict__ alpha_p, const float* __restrict__ bplus_p,
    const float* __restrict__ bminus_p, float* __restrict__ out,
    float* __restrict__ reg_loss)
{
  __shared__ __align__(16) float s_a_raw[HALO + LPAD + HALO];     // approx ping
  __shared__ __align__(16) float s_b_raw[HALO + LPAD / 2 + HALO]; // approx pong
  __shared__ __align__(16) float s_c[LPAD + 16];                  // coeffs + slack
  __shared__ float s_h[LEVELS * KTAPS];
  __shared__ float s_red[8];

  float* s_a = s_a_raw + HALO;
  float* s_b = s_b_raw + HALO;

  const int tid  = threadIdx.x;
  const int row  = blockIdx.x;
  const int lane = tid & 31;
  const int wid  = tid >> 5;
  const int half = lane >> 4;   // A: K-pair group 0/1.  D: M offset 0/8.
  const int mn   = lane & 15;   // A: M row.  B/D: N column.

  const float alpha = alpha_p[0], bp = bplus_p[0], bm = bminus_p[0];

  for (int i = tid; i < LEVELS * KTAPS; i += 256) s_h[i] = scaling[i];

  // zero permanent front halos + initial back halo of s_a
  if (tid < HALO) {
    s_a[tid - HALO] = 0.0f;
    s_b[tid - HALO] = 0.0f;
    s_a[LPAD + tid] = 0.0f;
  }

  // ---- Stage A: row -> LDS, bulk contiguous copy via TDM / async-LDS ----
  const float* xr = x + (size_t)row * M_IN;
  {
#if __has_builtin(__builtin_amdgcn_tensor_load_to_lds)
    if (wid == 0) {
      unsigned lds_off =
          (unsigned)(uintptr_t)(__attribute__((address_space(3))) const float*)(s_a + PL);
      u32x4 g0 = tdm_g0(xr, lds_off);
      i32x8 g1 = tdm_g1();
      i32x4 z4 = { 0, 0, 0, 0 };
#if __clang_major__ >= 23
      i32x8 z8 = { 0, 0, 0, 0, 0, 0, 0, 0 };
      __builtin_amdgcn_tensor_load_to_lds(g0, g1, z4, z4, z8, 0);
#else
      __builtin_amdgcn_tensor_load_to_lds(g0, g1, z4, z4, 0);
#endif
      __builtin_amdgcn_s_wait_tensorcnt(0);
    }
#elif __has_builtin(__builtin_amdgcn_global_load_async_to_lds_b128)
    const float4* x4 = (const float4*)xr;
    float4* a4 = (float4*)(s_a + PL);
    for (int i = tid; i < M_IN / 4; i += 256)
      __builtin_amdgcn_global_load_async_to_lds_b128(
          (gptr4)(i32x4*)(x4 + i), (lptr4)(i32x4*)(a4 + i), 0, 0);
    asm volatile("s_wait_asynccnt 0" ::: "memory");
#else
    const float4* x4 = (const float4*)xr;
    float4* a4 = (float4*)(s_a + PL);
    for (int i = tid; i < M_IN / 4; i += 256) a4[i] = x4[i];
#endif
    // reflect edges (192 each side)
    for (int p = tid; p < PL; p += 256) s_a[p] = xr[PL - p];
    for (int p = PL + M_IN + tid; p < LPAD; p += 256)
      s_a[p] = xr[2 * (M_IN - 1) - (p - PL)];
  }
  __syncthreads();

  // ---- Stage B: 12-level analysis via WMMA f32 16x16x4 ----
  int cur = 0;
  for (int lvl = 0; lvl < LEVELS; ++lvl) {
    const int n  = LPAD >> lvl;
    const int hn = n >> 1;
    const float* fin  = (cur == 0) ? s_a : s_b;
    float*       fout = (cur == 0) ? s_b : s_a;
    float*       det  = s_c + (LPAD - n);
    const float* hf   = s_h + lvl * KTAPS;

    // per-lane B operand: col 0 = h, col 1 = g (qmf: g[k]=(-1)^k h[15-k])
    float bx[4], by[4];
#pragma unroll
    for (int kg = 0; kg < 4; ++kg) {
      const int kb = kg * 4 + half * 2;
      const float h0 = hf[kb], h1 = hf[kb + 1];
      const float g0 = ((kb & 1) ? -1.0f : 1.0f) * hf[15 - kb];
      const float g1 = (((kb + 1) & 1) ? -1.0f : 1.0f) * hf[14 - kb];
      bx[kg] = (mn == 0) ? h0 : ((mn == 1) ? g0 : 0.0f);
      by[kg] = (mn == 0) ? h1 : ((mn == 1) ? g1 : 0.0f);
    }

    const int tiles = (hn + 15) >> 4;
    for (int t = wid; t < tiles; t += 8) {
      const int base = 32 * t + 2 * mn - 7 + half * 2;
      v8f c = {};
#pragma unroll
      for (int kg = 0; kg < 4; ++kg) {
        const int j = base + kg * 4;
        v2f A = { fin[j], fin[j + 1] };     // halo-backed: unconditional loads
        v2f B = { bx[kg], by[kg] };
        c = __builtin_amdgcn_wmma_f32_16x16x4_f32(false, A, false, B, (short)0, c,
                                                  false, false);
      }
      // Unconditional full-tile stores. Rows >= hn only occur on single-wave
      // levels; they land in slack / regions rewritten later, and this wave
      // re-zeroes the halo below (same-wave LDS ops are in order).
      if (mn <= 1) {
#pragma unroll
        for (int v = 0; v < 8; ++v) {
          const int idx = 16 * t + half * 8 + v;
          if (mn == 0) fout[idx] = c[v]; else det[idx] = c[v];
        }
      }
    }
    if (tid < HALO) fout[hn + tid] = 0.0f;  // back halo for next level
    cur ^= 1;
    __syncthreads();
  }
  if (tid < 4) s_c[LPAD - 4 + tid] = s_a[tid];  // final approx (cur==0 -> s_a)
  __syncthreads();

  // ---- Stage C: soft gate + L1 reg ----
  float lsum = 0.0f;
  for (int i = tid; i < LPAD; i += 256) {
    const float v  = s_c[i];
    const float fv = v * (sigm(alpha * (v - bp)) + sigm(-alpha * (v + bm)));
    s_c[i] = fv;
    lsum += fabsf(fv);
  }
  for (int off = 16; off > 0; off >>= 1) lsum += __shfl_down(lsum, off, 32);
  if (lane == 0) s_red[wid] = lsum;
  __syncthreads();
  if (tid == 0) {
    float s = 0.0f;
    for (int w = 0; w < 8; ++w) s += s_red[w];
    atomicAdd(reg_loss, s);
  }
  __syncthreads();

  // ---- Stage D: 12-level synthesis via WMMA (column 0) ----
  if (tid < 4)  s_a[tid] = s_c[LPAD - 4 + tid];
  if (tid >= 4 && tid < 4 + HALO) s_a[tid] = 0.0f;   // back halo of len-4 input
  __syncthreads();
  cur = 0;
  for (int lvl = LEVELS - 1; lvl >= 0; --lvl) {
    const int nl = LPAD >> (lvl + 1);   // input length
    const int on = nl << 1;             // output length
    const float* ain  = (cur == 0) ? s_a : s_b;
    float*       aout = (cur == 0) ? s_b : s_a;
    const float* det  = s_c + (LPAD - (LPAD >> lvl));
    const float* hf   = s_h + lvl * KTAPS;

    float bg0[4], bg1[4], bh0[4], bh1[4];
#pragma unroll
    for (int kg = 0; kg < 4; ++kg) {
      const int kb = kg * 4 + half * 2;
      const float g0 = ((kb & 1) ? -1.0f : 1.0f) * hf[15 - kb];
      const float g1 = (((kb + 1) & 1) ? -1.0f : 1.0f) * hf[14 - kb];
      bg0[kg] = (mn == 0) ? g0 : 0.0f;
      bg1[kg] = (mn == 0) ? g1 : 0.0f;
      bh0[kg] = (mn == 0) ? hf[kb]     : 0.0f;
      bh1[kg] = (mn == 0) ? hf[kb + 1] : 0.0f;
    }

    const int tiles = (on + 15) >> 4;
    for (int t = wid; t < tiles; t += 8) {
      const int jb = 16 * t + mn + 7;   // out[j] = sum_k f[k]*u[j+7-k]
      v8f c = {};
      // detail stream (packed in s_c: clamp address, branch-free select)
#pragma unroll
      for (int kg = 0; kg < 4; ++kg) {
        const int k0 = kg * 4 + half * 2;
        const int j0 = jb - k0, j1 = j0 - 1;
        const int i0 = j0 >> 1, i1 = j1 >> 1;
        const float v0 = det[iclamp(i0, 0, nl - 1)];
        const float v1 = det[iclamp(i1, 0, nl - 1)];
        const float a0 = (((j0 & 1) == 0) && ((unsigned)i0 < (unsigned)nl)) ? v0 : 0.0f;
        const float a1 = (((j1 & 1) == 0) && ((unsigned)i1 < (unsigned)nl)) ? v1 : 0.0f;
        v2f A = { a0, a1 };
        v2f B = { bg0[kg], bg1[kg] };
        c = __builtin_amdgcn_wmma_f32_16x16x4_f32(false, A, false, B, (short)0, c,
                                                  false, false);
      }
      // approx stream (halo-backed: unconditional loads, parity select only)
#pragma unroll
      for (int kg = 0; kg < 4; ++kg) {
        const int k0 = kg * 4 + half * 2;
        const int j0 = jb - k0, j1 = j0 - 1;
        const float v0 = ain[j0 >> 1];
        const float v1 = ain[j1 >> 1];
        const float a0 = ((j0 & 1) == 0) ? v0 : 0.0f;
        const float a1 = ((j1 & 1) == 0) ? v1 : 0.0f;
        v2f A = { a0, a1 };
        v2f B = { bh0[kg], bh1[kg] };
        c = __builtin_amdgcn_wmma_f32_16x16x4_f32(false, A, false, B, (short)0, c,
                                                  false, false);
      }
      if (mn == 0) {        // unconditional full-tile store (see analysis note)
#pragma unroll
        for (int v = 0; v < 8; ++v) aout[16 * t + half * 8 + v] = c[v];
      }
    }
    if (tid < HALO) aout[on + tid] = 0.0f;
    cur ^= 1;
    __syncthreads();
  }

  // ---- Stage E: crop + bulk store (final in s_a after 12 flips) ----
#if __has_builtin(__builtin_amdgcn_tensor_store_from_lds)
  if (wid == 0) {
    unsigned lds_off =
        (unsigned)(uintptr_t)(__attribute__((address_space(3))) const float*)(s_a + PL);
    u32x4 g0 = tdm_g0(out + (size_t)row * M_IN, lds_off);
    i32x8 g1 = tdm_g1();
    i32x4 z4 = { 0, 0, 0, 0 };
#if __clang_major__ >= 23
    i32x8 z8 = { 0, 0, 0, 0, 0, 0, 0, 0 };
    __builtin_amdgcn_tensor_store_from_lds(g0, g1, z4, z4, z8, 0);
#else
    __builtin_amdgcn_tensor_store_from_lds(g0, g1, z4, z4, 0);
#endif
    __builtin_amdgcn_s_wait_tensorcnt(0);
  }
#else
  const float4* r4 = (const float4*)(s_a + PL);
  float4* o4 = (float4*)(out + (size_t)row * M_IN);
  for (int i = tid; i < M_IN / 4; i += 256) o4[i] = r4[i];
#endif
}

extern "C" void kernel_launch(void* const* d_in, const int* in_sizes, int n_in,
                              void* d_out, int out_size, void* d_ws, size_t ws_size,
                              hipStream_t stream) {
  const float* x       = (const float*)d_in[0];
  const float* scaling = (const float*)d_in[1];
  const float* alpha   = (const float*)d_in[2];
  const float* bplus   = (const float*)d_in[3];
  const float* bminus  = (const float*)d_in[4];
  const int rows = in_sizes[0] / M_IN;      // 1024
  float* out = (float*)d_out;
  float* reg = out + (size_t)rows * M_IN;   // reg_loss scalar slot

  hipLaunchKernelGGL(Despawn_zero_kernel, dim3(1), dim3(1), 0, stream, reg);
  hipLaunchKernelGGL(Despawn_dwt_kernel, dim3(rows), dim3(256), 0, stream,
                     x, scaling, alpha, bplus, bminus, out, reg);
}